// PairwiseMultiRigModel_68839735820969
// MI455X (gfx1250) — compile-verified
//
#include <hip/hip_runtime.h>
#include <stdint.h>

// ---------------------------------------------------------------------------
// PairwiseMultiRigModel residual — MI455X (gfx1250) memory-bound gather kernel
//
//  * streamed read-once data + write-once output -> non-temporal (keep 192MB
//    L2 for the gather tables)
//  * gather tables (ref_rots / ref_trans / calib / points_3d) -> regular loads
//  * tiny rel_rots/rel_trans tables staged to LDS via gfx1250 async
//    global->LDS copy (global_load_async_to_lds_b32 / s_wait_asynccnt),
//    then per-point reads are ds_load_b128.
//  * all loop offsets are unsigned 32-bit so the backend can use the GVS
//    addressing mode (SGPR base + 32-bit VGPR offset) and skip 64-bit VALU
//    address math.
// ---------------------------------------------------------------------------

#define EPS_F 1e-8f

struct F3 { float x, y, z; };

__device__ __forceinline__ F3 cross3(const F3 a, const F3 b) {
    F3 r;
    r.x = fmaf(a.y, b.z, -(a.z * b.y));
    r.y = fmaf(a.z, b.x, -(a.x * b.z));
    r.z = fmaf(a.x, b.y, -(a.y * b.x));
    return r;
}

// qrot(q, x) = x + 2 * (qv x (qv x x + w*x))
__device__ __forceinline__ F3 qrot3(const F3 qv, const float w, const F3 x) {
    F3 c1 = cross3(qv, x);
    F3 t;
    t.x = fmaf(w, x.x, c1.x);
    t.y = fmaf(w, x.y, c1.y);
    t.z = fmaf(w, x.z, c1.z);
    F3 c2 = cross3(qv, t);
    F3 r;
    r.x = fmaf(2.0f, c2.x, x.x);
    r.y = fmaf(2.0f, c2.y, x.y);
    r.z = fmaf(2.0f, c2.z, x.z);
    return r;
}

__global__ __launch_bounds__(256) void pose_residual_kernel(
    const float*         __restrict__ feat,      // (N,3)   stream
    const int*           __restrict__ gidx,      // (N,2)   stream
    const int*           __restrict__ pidx,      // (N,)    stream
    const unsigned char* __restrict__ calib,     // (G,)    gather (bool, 1B)
    const float*         __restrict__ ref_rots,  // (G,4)   gather
    const float*         __restrict__ rel_rots,  // (16,4)  -> LDS
    const float*         __restrict__ points3d,  // (P,3)   gather
    const float*         __restrict__ ref_trans, // (G,3)   gather
    const float*         __restrict__ rel_trans, // (16,3)  -> LDS
    const float*         __restrict__ scales,    // (N,)    stream
    float*               __restrict__ out,       // (N,3)
    int n)
{
    // 16 members x {quat[4], trans[3], pad} = 128 floats, 32B record stride
    __shared__ __align__(16) float s_tab[16 * 8];

    const int tid = threadIdx.x;

    // ---- Stage rel tables into LDS with gfx1250 async global->LDS copies ----
    // 112 lanes each move one dword; per-lane LDS byte address in vdst VGPR,
    // 64-bit global address in vaddr pair (GV mode). Tracked by ASYNCcnt.
    if (tid < 112) {
        unsigned lds_off;
        unsigned long long ga;
        if (tid < 64) {
            const int m = tid >> 2, c = tid & 3;
            lds_off = (unsigned)(uintptr_t)(const void*)&s_tab[m * 8 + c];
            ga      = (unsigned long long)(uintptr_t)(const void*)(rel_rots + tid);
        } else {
            const int j = tid - 64;
            const int m = j / 3, c = j - 3 * m;
            lds_off = (unsigned)(uintptr_t)(const void*)&s_tab[m * 8 + 4 + c];
            ga      = (unsigned long long)(uintptr_t)(const void*)(rel_trans + j);
        }
        asm volatile("global_load_async_to_lds_b32 %0, %1, off"
                     :: "v"(lds_off), "v"(ga) : "memory");
    }
    asm volatile("s_wait_asynccnt 0" ::: "memory");
    __syncthreads();

    const unsigned stride = gridDim.x * blockDim.x;   // <= 1M, fits 32-bit
    for (unsigned i = blockIdx.x * blockDim.x + (unsigned)tid; i < (unsigned)n;
         i += stride) {
        const unsigned i2 = 2u * i;   // <= 8.4M elems, byte offsets < 2^31
        const unsigned i3 = 3u * i;

        // ---- streamed inputs: read exactly once -> non-temporal ----
        const int   g   = __builtin_nontemporal_load(gidx + i2);
        const int   m   = __builtin_nontemporal_load(gidx + i2 + 1u);
        const int   pid = __builtin_nontemporal_load(pidx + i);
        const float sc  = __builtin_nontemporal_load(scales + i);
        F3 f;
        f.x = __builtin_nontemporal_load(feat + i3);
        f.y = __builtin_nontemporal_load(feat + i3 + 1u);
        f.z = __builtin_nontemporal_load(feat + i3 + 2u);

        const unsigned g4 = 4u * (unsigned)g;
        const unsigned g3 = 3u * (unsigned)g;
        const unsigned p3o = 3u * (unsigned)pid;

        // ---- table gathers: regular loads, stay L2 / WGP$ resident ----
        const float4 fq = *(const float4*)(ref_rots + g4);
        F3 ft;
        ft.x = ref_trans[g3];
        ft.y = ref_trans[g3 + 1u];
        ft.z = ref_trans[g3 + 2u];
        const bool cal = calib[(unsigned)g] != 0;
        F3 p3;
        p3.x = points3d[p3o];
        p3.y = points3d[p3o + 1u];
        p3.z = points3d[p3o + 2u];

        // ---- rel tables from LDS (ds_load_b128, 16B aligned) ----
        const float4 rq  = *(const float4*)&s_tab[8 * m];
        const float4 rt4 = *(const float4*)&s_tab[8 * m + 4];

        const F3 rv = { rq.x, rq.y, rq.z };  const float rw = rq.w;
        const F3 fv = { fq.x, fq.y, fq.z };  const float fw = fq.w;

        // pose_R = qmul(rel_R, ref_R)
        const F3 cx = cross3(rv, fv);
        F3 pv;
        pv.x = fmaf(rw, fv.x, fmaf(fw, rv.x, cx.x));
        pv.y = fmaf(rw, fv.y, fmaf(fw, rv.y, cx.y));
        pv.z = fmaf(rw, fv.z, fmaf(fw, rv.z, cx.z));
        const float pw = fmaf(rw, fw,
                         -fmaf(rv.x, fv.x, fmaf(rv.y, fv.y, rv.z * fv.z)));

        // pose_t = -qrot(conj(ref_R), qrot(conj(rel_R), rel_t) + ref_t)
        const F3 nrv = { -rv.x, -rv.y, -rv.z };
        const F3 rt  = { rt4.x, rt4.y, rt4.z };
        const F3 inner = qrot3(nrv, rw, rt);
        const F3 s = { inner.x + ft.x, inner.y + ft.y, inner.z + ft.z };
        const F3 nfv = { -fv.x, -fv.y, -fv.z };
        const F3 outer = qrot3(nfv, fw, s);   // pose_t == -outer

        // translations = qrot(conj(pose_R), feature)
        const F3 npv = { -pv.x, -pv.y, -pv.z };
        const F3 tr = qrot3(npv, pw, f);

        const float d   = fmaf(tr.x, tr.x, fmaf(tr.y, tr.y, tr.z * tr.z));
        const float nrm = sqrtf(d);
        const float mul = cal ? 1.0f : (1.0f / (nrm + EPS_F));
        const float sm  = sc * mul;

        // residual = sm*tr - (p3 - pose_t) = sm*tr - p3 - outer
        const float rx = fmaf(sm, tr.x, -(p3.x + outer.x));
        const float ry = fmaf(sm, tr.y, -(p3.y + outer.y));
        const float rz = fmaf(sm, tr.z, -(p3.z + outer.z));

        // ---- write-once output -> non-temporal store ----
        __builtin_nontemporal_store(rx, out + i3);
        __builtin_nontemporal_store(ry, out + i3 + 1u);
        __builtin_nontemporal_store(rz, out + i3 + 2u);
    }
}

extern "C" void kernel_launch(void* const* d_in, const int* in_sizes, int n_in,
                              void* d_out, int out_size, void* d_ws, size_t ws_size,
                              hipStream_t stream) {
    (void)n_in; (void)d_ws; (void)ws_size; (void)out_size;
    const float*         feat      = (const float*)d_in[0];
    const int*           gidx      = (const int*)d_in[1];
    const int*           pidx      = (const int*)d_in[2];
    const unsigned char* calib     = (const unsigned char*)d_in[3]; // jnp bool_ -> 1 byte
    const float*         ref_rots  = (const float*)d_in[4];
    const float*         rel_rots  = (const float*)d_in[5];
    const float*         points3d  = (const float*)d_in[6];
    const float*         ref_trans = (const float*)d_in[7];
    const float*         rel_trans = (const float*)d_in[8];
    const float*         scales    = (const float*)d_in[9];
    float*               out       = (float*)d_out;

    const int n = in_sizes[2];          // point_indices count == N
    const int block = 256;              // 8 wave32s per workgroup
    long long need = ((long long)n + block - 1) / block;
    int grid = (need > 4096) ? 4096 : (int)need;   // grid-stride amortizes LDS staging
    if (grid < 1) grid = 1;

    pose_residual_kernel<<<grid, block, 0, stream>>>(
        feat, gidx, pidx, calib, ref_rots, rel_rots, points3d,
        ref_trans, rel_trans, scales, out, n);
}